// MulticastAwareGNN_25469156065923
// MI455X (gfx1250) — compile-verified
//
#include <hip/hip_runtime.h>
#include <hip/hip_bf16.h>
#include <math.h>

typedef float v2f __attribute__((ext_vector_type(2)));
typedef float v8f __attribute__((ext_vector_type(8)));

#define HDIM   128
#define HEADS  4
#define HD     32
#define NEGSL  0.2f

// ---------- helpers ----------
__device__ __forceinline__ unsigned enc_ord(float f) {
  unsigned i = __float_as_uint(f);
  return (i & 0x80000000u) ? ~i : (i | 0x80000000u);
}
__device__ __forceinline__ float dec_ord(unsigned u) {
  return (u & 0x80000000u) ? __uint_as_float(u ^ 0x80000000u)
                           : __uint_as_float(~u);
}
__device__ __forceinline__ void atomAddF(float* p, float v) {
  unsafeAtomicAdd(p, v);  // native global_atomic_add_f32 on gfx1250
}

// ---------- K_sa: segment-sum raw edge_attr by dst (for self-loop fill) ----------
__global__ void k_sa(const int* __restrict__ ei, const float* __restrict__ ea,
                     float* __restrict__ sa, float* __restrict__ cnt, int E) {
  int e = blockIdx.x * blockDim.x + threadIdx.x;
  if (e >= E) return;
  int d = ei[E + e];
  #pragma unroll
  for (int k = 0; k < 8; ++k) atomAddF(&sa[(size_t)d * 8 + k], ea[(size_t)e * 8 + k]);
  atomAddF(&cnt[d], 1.0f);
}

// ---------- K_embed: h = x @ Wn + bn ----------
__global__ void k_embed(const float* __restrict__ x, const float* __restrict__ Wn,
                        const float* __restrict__ bn, float* __restrict__ h, int N) {
  int n = blockIdx.x;
  int c = threadIdx.x;
  if (n >= N) return;
  const float* xr = x + (size_t)n * 16;
  float acc = bn[c];
  #pragma unroll
  for (int k = 0; k < 16; ++k) acc = fmaf(xr[k], Wn[k * HDIM + c], acc);
  h[(size_t)n * HDIM + c] = acc;
}

// ---------- K_prep: per-layer folded edge-attention matrices ----------
// v[k][h]   = sum_d W_edge[l][k][h*32+d] * att_e[l][h][d]      (128x4)
// Mv[l][k8][h] = sum_c We[k8][c] * v[c][h]                      (8x4)
// cv[l][h]     = sum_c be[c]    * v[c][h]                       (4)
__global__ void k_prep(const float* __restrict__ W_edge, const float* __restrict__ att_e,
                       const float* __restrict__ We, const float* __restrict__ be,
                       float* __restrict__ Mv, float* __restrict__ cv) {
  __shared__ float V[HDIM][HEADS];
  int l = blockIdx.x;
  int t = threadIdx.x;  // 128 threads
  const float* Wl = W_edge + (size_t)l * HDIM * HDIM;
  const float* Al = att_e + (size_t)l * HEADS * HD;
  #pragma unroll
  for (int hh = 0; hh < HEADS; ++hh) {
    float s = 0.f;
    #pragma unroll
    for (int d = 0; d < HD; ++d) s = fmaf(Wl[t * HDIM + hh * HD + d], Al[hh * HD + d], s);
    V[t][hh] = s;
  }
  __syncthreads();
  if (t < 32) {
    int k = t >> 2, hh = t & 3;
    float s = 0.f;
    for (int c = 0; c < HDIM; ++c) s = fmaf(We[k * HDIM + c], V[c][hh], s);
    Mv[l * 32 + k * 4 + hh] = s;
  }
  if (t < 4) {
    float s = 0.f;
    for (int c = 0; c < HDIM; ++c) s = fmaf(be[c], V[c][t], s);
    cv[l * 4 + t] = s;
  }
}

// ---------- K_ael: self-loop attention-edge scores per layer ----------
// aEL[l][n][h] = (sum_k sa[n][k]*Mv[l][k][h] + cnt[n]*cv[l][h]) / max(cnt[n],1)
__global__ void k_ael(const float* __restrict__ sa, const float* __restrict__ cnt,
                      const float* __restrict__ Mv, const float* __restrict__ cv,
                      float* __restrict__ aEL, int N) {
  int n = blockIdx.x * blockDim.x + threadIdx.x;
  int l = blockIdx.y;
  if (n >= N) return;
  float c = cnt[n];
  float inv = 1.0f / fmaxf(c, 1.0f);
  float s8[8];
  #pragma unroll
  for (int k = 0; k < 8; ++k) s8[k] = sa[(size_t)n * 8 + k];
  #pragma unroll
  for (int hh = 0; hh < HEADS; ++hh) {
    float s = c * cv[l * 4 + hh];
    #pragma unroll
    for (int k = 0; k < 8; ++k) s = fmaf(s8[k], Mv[l * 32 + k * 4 + hh], s);
    aEL[((size_t)l * N + n) * 4 + hh] = s * inv;
  }
}

// ---------- K3: LN + xs = LN(h) @ Wg  (fp32 WMMA) + a_src/a_dst ----------
__global__ void __launch_bounds__(128)
k_ln_gemm_attn(const float* __restrict__ h, const float* __restrict__ Wg,
               const float* __restrict__ lng, const float* __restrict__ lnb,
               const float* __restrict__ att_s, const float* __restrict__ att_d,
               float* __restrict__ xs, float* __restrict__ a_s, float* __restrict__ a_d,
               int N) {
  __shared__ float X[16][132];        // padded: bank = (4*row + k) % 64, conflict-free
  __shared__ float red[16][8][2];
  __shared__ float MU[16], RS[16];
  __shared__ float AS[16][HEADS], AD[16][HEADS];

  int t = threadIdx.x;
  int nodeBase = blockIdx.x * 16;
  int row = t >> 3, seg = t & 7;      // 8 threads per row, 16 cols each
  int n = nodeBase + row; if (n >= N) n = N - 1;

  float vals[16];
  float s0 = 0.f, s1 = 0.f;
  const float* hr = h + (size_t)n * HDIM + seg * 16;
  #pragma unroll
  for (int i = 0; i < 16; ++i) { float v = hr[i]; vals[i] = v; s0 += v; s1 += v * v; }
  red[row][seg][0] = s0; red[row][seg][1] = s1;
  if (t < 64) { AS[t >> 2][t & 3] = 0.f; AD[t >> 2][t & 3] = 0.f; }
  __syncthreads();
  if (seg == 0) {
    float m = 0.f, q = 0.f;
    #pragma unroll
    for (int i = 0; i < 8; ++i) { m += red[row][i][0]; q += red[row][i][1]; }
    m *= (1.f / HDIM);
    q = q * (1.f / HDIM) - m * m;
    MU[row] = m; RS[row] = rsqrtf(q + 1e-5f);
  }
  __syncthreads();
  float mu = MU[row], rs = RS[row];
  #pragma unroll
  for (int i = 0; i < 16; ++i) {
    int c = seg * 16 + i;
    X[row][c] = (vals[i] - mu) * rs * lng[c] + lnb[c];
  }
  __syncthreads();

  // GEMM: wave w owns head w -> output cols [32w, 32w+32) = two 16x16 tiles
  int wv = t >> 5, lane = t & 31;
  int colBase = wv * 32;
  int la = lane & 15;
  int koff = (lane >> 4) * 2;         // A/B fragment K sub-offset per half-wave
  v8f acc0 = {}; v8f acc1 = {};
  #pragma unroll 4
  for (int k4 = 0; k4 < 32; ++k4) {
    int k = k4 * 4 + koff;
    v2f a; a.x = X[la][k]; a.y = X[la][k + 1];
    const float* Bp = Wg + (size_t)k * HDIM + colBase + la;
    v2f b0; b0.x = Bp[0];  b0.y = Bp[HDIM];
    v2f b1; b1.x = Bp[16]; b1.y = Bp[HDIM + 16];
    acc0 = __builtin_amdgcn_wmma_f32_16x16x4_f32(false, a, false, b0, (short)0, acc0, false, false);
    acc1 = __builtin_amdgcn_wmma_f32_16x16x4_f32(false, a, false, b1, (short)0, acc1, false, false);
  }

  // store D + fold attention dot products (head == wv)
  float ws0 = att_s[wv * HD + la], ws1 = att_s[wv * HD + 16 + la];
  float wd0 = att_d[wv * HD + la], wd1 = att_d[wv * HD + 16 + la];
  int rbase = (lane >> 4) * 8;
  #pragma unroll
  for (int r = 0; r < 8; ++r) {
    int rr = rbase + r;
    int nn = nodeBase + rr;
    float d0 = acc0[r], d1 = acc1[r];
    if (nn < N) {
      xs[(size_t)nn * HDIM + colBase + la]      = d0;
      xs[(size_t)nn * HDIM + colBase + 16 + la] = d1;
    }
    atomicAdd(&AS[rr][wv], d0 * ws0 + d1 * ws1);   // ds_add_f32
    atomicAdd(&AD[rr][wv], d0 * wd0 + d1 * wd1);
  }
  __syncthreads();
  if (t < 64) {
    int rr = t >> 2, hh = t & 3;
    int nn = nodeBase + rr;
    if (nn < N) {
      a_s[(size_t)nn * 4 + hh] = AS[rr][hh];
      a_d[(size_t)nn * 4 + hh] = AD[rr][hh];
    }
  }
}

// ---------- K4: alpha = leaky_relu(a_s[src]+a_d[dst]+a_edge); segment max ----------
__global__ void k_attn_edge(const int* __restrict__ ei, const float* __restrict__ ea,
                            const float* __restrict__ a_s, const float* __restrict__ a_d,
                            const float* __restrict__ aELl,
                            const float* __restrict__ Mv, const float* __restrict__ cv,
                            float* __restrict__ alpha, unsigned* __restrict__ mx,
                            int E, int N) {
  int idx = blockIdx.x * blockDim.x + threadIdx.x;
  int T = E + N;
  if (idx >= T) return;
  int s_, d_;
  float ae[HEADS];
  if (idx < E) {
    s_ = ei[idx]; d_ = ei[E + idx];
    float e8[8];
    #pragma unroll
    for (int k = 0; k < 8; ++k) e8[k] = ea[(size_t)idx * 8 + k];
    #pragma unroll
    for (int hh = 0; hh < HEADS; ++hh) {
      float s = cv[hh];
      #pragma unroll
      for (int k = 0; k < 8; ++k) s = fmaf(e8[k], Mv[k * 4 + hh], s);
      ae[hh] = s;
    }
  } else {
    int n0 = idx - E; s_ = n0; d_ = n0;
    #pragma unroll
    for (int hh = 0; hh < HEADS; ++hh) ae[hh] = aELl[(size_t)n0 * 4 + hh];
  }
  #pragma unroll
  for (int hh = 0; hh < HEADS; ++hh) {
    float a = a_s[(size_t)s_ * 4 + hh] + a_d[(size_t)d_ * 4 + hh] + ae[hh];
    a = (a > 0.f) ? a : NEGSL * a;
    alpha[(size_t)idx * 4 + hh] = a;
    atomicMax(&mx[(size_t)d_ * 4 + hh], enc_ord(a));
  }
}

// ---------- K5: ex = exp(alpha - max); segment sum ----------
__global__ void k_softmax_den(const int* __restrict__ ei, float* __restrict__ alpha,
                              const unsigned* __restrict__ mx, float* __restrict__ ssum,
                              int E, int N) {
  int idx = blockIdx.x * blockDim.x + threadIdx.x;
  int T = E + N;
  if (idx >= T) return;
  int d_ = (idx < E) ? ei[E + idx] : (idx - E);
  #pragma unroll
  for (int hh = 0; hh < HEADS; ++hh) {
    float m = dec_ord(mx[(size_t)d_ * 4 + hh]);
    float x = __expf(alpha[(size_t)idx * 4 + hh] - m);
    alpha[(size_t)idx * 4 + hh] = x;
    atomAddF(&ssum[(size_t)d_ * 4 + hh], x);
  }
}

// ---------- K6: out[dst] += (ex/sum[dst]) * xs[src] ----------
__global__ void __launch_bounds__(256)
k_aggregate(const int* __restrict__ ei, const float* __restrict__ alpha,
            const float* __restrict__ ssum, const float* __restrict__ xs,
            float* __restrict__ outb, int E, int N) {
  int t = threadIdx.x;
  long long idx = (long long)blockIdx.x * 2 + (t >> 7);
  int col = t & 127;
  long long T = (long long)E + N;
  if (idx >= T) return;
  int s_, d_;
  if (idx < E) { s_ = ei[idx]; d_ = ei[E + idx]; }
  else         { s_ = (int)(idx - E); d_ = s_; }
  int hh = col >> 5;
  float w = alpha[idx * 4 + hh] / ssum[(size_t)d_ * 4 + hh];
  atomAddF(&outb[(size_t)d_ * HDIM + col], w * xs[(size_t)s_ * HDIM + col]);
}

// ---------- K7: h += relu(out + bg) ----------
__global__ void k_update(float* __restrict__ h, const float* __restrict__ outb,
                         const float* __restrict__ bgl, int N) {
  long long idx = (long long)blockIdx.x * blockDim.x + threadIdx.x;
  if (idx >= (long long)N * HDIM) return;
  int c = (int)(idx & (HDIM - 1));
  float o = outb[idx] + bgl[c];
  h[idx] += fmaxf(o, 0.f);
}

// ---------- K8: graph_emb = mean over nodes ----------
__global__ void k_mean(const float* __restrict__ h, float* __restrict__ emb, int N) {
  int c = threadIdx.x;
  int r0 = blockIdx.x * 128;
  int r1 = min(r0 + 128, N);
  float s = 0.f;
  for (int r = r0; r < r1; ++r) s += h[(size_t)r * HDIM + c];
  atomAddF(&emb[c], s * (1.0f / (float)N));
}

// ---------- host ----------
extern "C" void kernel_launch(void* const* d_in, const int* in_sizes, int n_in,
                              void* d_out, int out_size, void* d_ws, size_t ws_size,
                              hipStream_t stream) {
  const float* x      = (const float*)d_in[0];
  const int*   ei     = (const int*)  d_in[1];
  const float* ea     = (const float*)d_in[2];
  const float* Wn     = (const float*)d_in[3];
  const float* bn     = (const float*)d_in[4];
  const float* We     = (const float*)d_in[5];
  const float* be     = (const float*)d_in[6];
  const float* ln_g   = (const float*)d_in[7];
  const float* ln_b   = (const float*)d_in[8];
  const float* Wg     = (const float*)d_in[9];
  const float* att_s  = (const float*)d_in[10];
  const float* att_d  = (const float*)d_in[11];
  const float* att_e  = (const float*)d_in[12];
  const float* W_edge = (const float*)d_in[13];
  const float* bg     = (const float*)d_in[14];

  int N = in_sizes[0] / 16;
  int E = in_sizes[2] / 8;
  long long T = (long long)E + N;

  float* h   = (float*)d_out;             // [N,128] output lives here
  float* emb = h + (size_t)N * HDIM;      // [1,128]

  float* ws = (float*)d_ws;
  size_t o = 0;
  float* xs    = ws + o; o += (size_t)N * HDIM;
  float* a_s   = ws + o; o += (size_t)N * 4;
  float* a_d   = ws + o; o += (size_t)N * 4;
  float* sa    = ws + o; o += (size_t)N * 8;   // [sa|cnt] contiguous -> one memset
  float* cnt   = ws + o; o += (size_t)N;
  float* aEL   = ws + o; o += (size_t)3 * N * 4;
  float* alpha = ws + o; o += (size_t)T * 4;
  unsigned* mx = (unsigned*)(ws + o); o += (size_t)N * 4;  // [mx|ssum|out] contiguous
  float* ssum  = ws + o; o += (size_t)N * 4;
  float* outb  = ws + o; o += (size_t)N * HDIM;
  float* Mv    = ws + o; o += 96;
  float* cv    = ws + o; o += 16;

  // self-loop fill precompute
  hipMemsetAsync(sa, 0, (size_t)N * 9 * sizeof(float), stream);
  k_sa<<<(E + 255) / 256, 256, 0, stream>>>(ei, ea, sa, cnt, E);
  // node embedding
  k_embed<<<N, HDIM, 0, stream>>>(x, Wn, bn, h, N);
  // folded edge-attention matrices + self-loop scores, all 3 layers
  k_prep<<<3, 128, 0, stream>>>(W_edge, att_e, We, be, Mv, cv);
  dim3 gael((N + 255) / 256, 3);
  k_ael<<<gael, 256, 0, stream>>>(sa, cnt, Mv, cv, aEL, N);

  for (int l = 0; l < 3; ++l) {
    hipMemsetAsync(mx, 0, (size_t)N * (4 + 4 + HDIM) * sizeof(float), stream);
    k_ln_gemm_attn<<<(N + 15) / 16, 128, 0, stream>>>(
        h, Wg + (size_t)l * HDIM * HDIM, ln_g + l * HDIM, ln_b + l * HDIM,
        att_s + l * HEADS * HD, att_d + l * HEADS * HD, xs, a_s, a_d, N);
    k_attn_edge<<<(int)((T + 255) / 256), 256, 0, stream>>>(
        ei, ea, a_s, a_d, aEL + (size_t)l * N * 4, Mv + l * 32, cv + l * 4,
        alpha, mx, E, N);
    k_softmax_den<<<(int)((T + 255) / 256), 256, 0, stream>>>(ei, alpha, mx, ssum, E, N);
    k_aggregate<<<(int)((T + 1) / 2), 256, 0, stream>>>(ei, alpha, ssum, xs, outb, E, N);
    k_update<<<(int)(((long long)N * HDIM + 255) / 256), 256, 0, stream>>>(
        h, outb, bg + l * HDIM, N);
  }

  hipMemsetAsync(emb, 0, HDIM * sizeof(float), stream);
  k_mean<<<(N + 127) / 128, 128, 0, stream>>>(h, emb, N);
}